// LennardJonesPurePyTorch_43937515438568
// MI455X (gfx1250) — compile-verified
//
#include <hip/hip_runtime.h>

// ---------------------------------------------------------------------------
// Lennard-Jones per-atom energy, half neighbor list (MI455X / gfx1250).
//   e_edge = 4*eps*((sig/r)^12 - (sig/r)^6) - SHIFT ; eps = sig = 1
//   out[i] += 0.5*e ; out[j] += 0.5*e   (atomic scatter)
//
// Roofline: 3.2M edges x 20 B = 64 MB streamed once -> ~2.7 us @ 23.3 TB/s.
// ~10 FLOP/edge -> compute negligible. Output = 400 KB, resident in 192 MB L2,
// so the 6.4M scatter adds run at L2 atomic rate. No data reuse / no matmul
// structure -> WMMA inapplicable; the gfx1250 path is b128 NT loads + native
// non-returning global_atomic_add_f32 at device scope (inline asm so the
// compiler cannot fall back to a CAS retry loop).
// ---------------------------------------------------------------------------

typedef float f32x4 __attribute__((ext_vector_type(4)));
typedef int   i32x4 __attribute__((ext_vector_type(4)));

#define LJ_SHIFT (-2.55983616e-4f)   // 4*((1/5)^12 - (1/5)^6)

__global__ void lj_zero_kernel(float* __restrict__ out, int n) {
    int i = blockIdx.x * blockDim.x + threadIdx.x;
    if (i < n) out[i] = 0.0f;
}

__device__ __forceinline__ float lj_half_energy(float x, float y, float z) {
    float r2  = __builtin_fmaf(x, x, __builtin_fmaf(y, y, z * z));
    float inv = 1.0f / r2;                 // (sigma/r)^2
    float s6  = inv * inv * inv;           // (sigma/r)^6
    float s12 = s6 * s6;                   // (sigma/r)^12
    float e   = 4.0f * (s12 - s6) - LJ_SHIFT;
    return 0.5f * e;
}

__device__ __forceinline__ void lj_scatter(float* __restrict__ out, int idx, float h) {
    // Native CDNA5 f32 atomic add: non-returning (STOREcnt, no wait, no VGPR
    // writeback), device scope so the RMW is performed coherently in L2.
    float* p = out + idx;
    asm volatile("global_atomic_add_f32 %0, %1, off scope:SCOPE_DEV"
                 :
                 : "v"(p), "v"(h)
                 : "memory");
}

__global__ __launch_bounds__(256)
void lj_edges_kernel(const float* __restrict__ dist,
                     const int*   __restrict__ ai,
                     const int*   __restrict__ aj,
                     float*       __restrict__ out,
                     int n_edges) {
    long t  = (long)blockIdx.x * blockDim.x + threadIdx.x;
    long e0 = t * 4;

    if (e0 + 3 < (long)n_edges) {
        // 4 edges per lane: 48 B of distances (3 x global_load_b128, 16B
        // aligned since 48B/thread stride), 16 B of each index array (b128).
        // All loads non-temporal: the 64 MB one-shot stream must not evict
        // the L2-resident 400 KB accumulator.
        const f32x4* dp = (const f32x4*)(dist + e0 * 3);
        f32x4 a = __builtin_nontemporal_load(dp + 0);   // e0.xyz,  e1.x
        f32x4 b = __builtin_nontemporal_load(dp + 1);   // e1.yz,   e2.xy
        f32x4 c = __builtin_nontemporal_load(dp + 2);   // e2.z,    e3.xyz
        i32x4 iv = __builtin_nontemporal_load((const i32x4*)(ai + e0));
        i32x4 jv = __builtin_nontemporal_load((const i32x4*)(aj + e0));

        float h0 = lj_half_energy(a.x, a.y, a.z);
        float h1 = lj_half_energy(a.w, b.x, b.y);
        float h2 = lj_half_energy(b.z, b.w, c.x);
        float h3 = lj_half_energy(c.y, c.z, c.w);

        lj_scatter(out, iv.x, h0);
        lj_scatter(out, jv.x, h0);
        lj_scatter(out, iv.y, h1);
        lj_scatter(out, jv.y, h1);
        lj_scatter(out, iv.z, h2);
        lj_scatter(out, jv.z, h2);
        lj_scatter(out, iv.w, h3);
        lj_scatter(out, jv.w, h3);
    } else {
        // Tail (n_edges not divisible by 4) — scalar path.
        for (long e = e0; e < (long)n_edges; ++e) {
            float x = dist[e * 3 + 0];
            float y = dist[e * 3 + 1];
            float z = dist[e * 3 + 2];
            float h = lj_half_energy(x, y, z);
            lj_scatter(out, ai[e], h);
            lj_scatter(out, aj[e], h);
        }
    }
}

extern "C" void kernel_launch(void* const* d_in, const int* in_sizes, int n_in,
                              void* d_out, int out_size, void* d_ws, size_t ws_size,
                              hipStream_t stream) {
    const float* dist = (const float*)d_in[0];   // [E,3] f32
    const int*   ai   = (const int*)d_in[1];     // [E]   i32
    const int*   aj   = (const int*)d_in[2];     // [E]   i32
    float*       out  = (float*)d_out;           // [n_atoms] f32
    const int n_edges = in_sizes[1];

    // Zero the accumulator every call (harness poisons d_out once; graph
    // replays must be deterministic). Same stream -> ordered before edges.
    {
        int blocks = (out_size + 255) / 256;
        lj_zero_kernel<<<blocks, 256, 0, stream>>>(out, out_size);
    }

    // 4 edges per thread.
    {
        int n_threads = (n_edges + 3) / 4;
        int blocks = (n_threads + 255) / 256;
        lj_edges_kernel<<<blocks, 256, 0, stream>>>(dist, ai, aj, out, n_edges);
    }
}